// lossYoloV2_59416577573082
// MI455X (gfx1250) — compile-verified
//
#include <hip/hip_runtime.h>
#include <hip/hip_bf16.h>

#define BB      256
#define NUM     5
#define CLASSES 20
#define HH      19
#define WW      19
#define AL      (CLASSES + 5)   // 25
#define HW      (HH * WW)       // 361
#define NT      50
#define THRESH  0.5f

typedef __attribute__((ext_vector_type(2))) float v2f;
typedef __attribute__((ext_vector_type(8))) float v8f;

// ---------------------------------------------------------------------------
// Wave32 sum-reduction using V_WMMA_F32_16X16X4_F32 (codegen-verified round 1).
// A = ones(16x4)  =>  D[m,n] = sum_k B[k,n].  The 32 lane partials occupy 32
// distinct B slots (VGPR1 of B = 0); every row of D is the identical vector of
// 16 column sums, so d.s0 of lane L = colsum(L % 16). Four xor-shuffles sum
// the 16 distinct column sums -> full 32-lane total in every lane.
// ---------------------------------------------------------------------------
__device__ __forceinline__ float wave_reduce_sum(float v) {
#if defined(__gfx1250__) && __has_builtin(__builtin_amdgcn_wmma_f32_16x16x4_f32)
  v2f a; a.x = 1.0f; a.y = 1.0f;   // A = all ones
  v2f b; b.x = v;    b.y = 0.0f;   // lane partial into one B slot
  v8f c = {};
  v8f d = __builtin_amdgcn_wmma_f32_16x16x4_f32(
      /*neg_a=*/false, a, /*neg_b=*/false, b,
      /*c_mod=*/(short)0, c, /*reuse_a=*/false, /*reuse_b=*/false);
  float s = d.s0;                  // column sum for this lane's column
  s += __shfl_xor(s, 1, 32);
  s += __shfl_xor(s, 2, 32);
  s += __shfl_xor(s, 4, 32);
  s += __shfl_xor(s, 8, 32);
  return s;
#else
  for (int m = 16; m >= 1; m >>= 1) v += __shfl_xor(v, m, 32);
  return v;
#endif
}

// Result valid in thread 0. Must be called from uniform control flow
// (blockDim.x a multiple of 32) so EXEC is all ones at the WMMA.
__device__ __forceinline__ float block_reduce_sum(float v, float* lds) {
  __syncthreads();                       // lds may be reused across calls
  const int lane = threadIdx.x & 31;
  const int wv   = threadIdx.x >> 5;
  float s = wave_reduce_sum(v);
  if (lane == 0) lds[wv] = s;
  __syncthreads();
  float tot = 0.f;
  if (threadIdx.x == 0) {
    const int nw = (blockDim.x + 31) >> 5;
    for (int i = 0; i < nw; ++i) tot += lds[i];
  }
  return tot;
}

__device__ __forceinline__ float box_iou(float acx, float acy, float aw, float ah,
                                         float bcx, float bcy, float bw, float bh) {
  float ax1 = acx - 0.5f * aw, ay1 = acy - 0.5f * ah;
  float ax2 = acx + 0.5f * aw, ay2 = acy + 0.5f * ah;
  float bx1 = bcx - 0.5f * bw, by1 = bcy - 0.5f * bh;
  float bx2 = bcx + 0.5f * bw, by2 = bcy + 0.5f * bh;
  float iw = fminf(ax2, bx2) - fmaxf(ax1, bx1);
  float ih = fminf(ay2, by2) - fmaxf(ay1, by1);
  if (iw < 0.f || ih < 0.f) return 0.f;
  float inter = iw * ih;
  float uni   = aw * ah + bw * bh - inter;
  return inter / uni;
}

// ---------------------------------------------------------------------------
// Kernel A: one block per image. loss_noobj partial + warm partial.
// Truth box extents/areas hoisted into LDS (bit-identical expressions, just
// computed once per block instead of per IoU).
// ---------------------------------------------------------------------------
__global__ void __launch_bounds__(384)
k_noobj_warm(const float* __restrict__ x, const float* __restrict__ truth,
             float* __restrict__ ws_noobj, float* __restrict__ ws_warm) {
  __shared__ float bx1[NT], by1[NT], bx2[NT], by2[NT], barea[NT];
  __shared__ int   s_nv;
  __shared__ float s_tw[NT];
  __shared__ float red[16];

  const int b   = blockIdx.x;
  const int tid = threadIdx.x;

  if (tid < NT) {
    const float* tr = truth + (size_t)(b * NT + tid) * 5;
    const float txv = tr[0], tyv = tr[1], twv = tr[2], thv = tr[3];
    s_tw[tid] = twv;
    bx1[tid] = txv - 0.5f * twv;  bx2[tid] = txv + 0.5f * twv;
    by1[tid] = tyv - 0.5f * thv;  by2[tid] = tyv + 0.5f * thv;
    barea[tid] = twv * thv;
  }
  __syncthreads();
  if (tid == 0) {   // validity = cumprod(w >= 1e-5)  -> prefix length
    int nv = 0;
    while (nv < NT && s_tw[nv] >= 1e-5f) ++nv;
    s_nv = nv;
  }
  __syncthreads();
  const int nv = s_nv;

  float noobj_acc = 0.f;
  float warm_acc  = 0.f;

  for (int cell = tid; cell < HW; cell += blockDim.x) {
    const float obj = x[((size_t)b * (NUM * AL) + 4) * HW + cell];
    int cnt = 0;
    for (int n = 0; n < NUM; ++n) {
      const float* px = x + ((size_t)(b * NUM + n) * AL) * HW + cell;
      const float ax = px[0 * HW], ay = px[1 * HW];
      const float aw = px[2 * HW], ah = px[3 * HW];
      // warm term ((tgt - pred) * 0.01)^2, tgt = {0.5, 0.5, 0, 0}
      float d0 = (0.5f - ax) * 0.01f, d1 = (0.5f - ay) * 0.01f;
      float d2 = (0.0f - aw) * 0.01f, d3 = (0.0f - ah) * 0.01f;
      warm_acc += d0 * d0 + d1 * d1 + d2 * d2 + d3 * d3;
      // pred box extents (loop-invariant over truths)
      const float ax1 = ax - 0.5f * aw, ay1 = ay - 0.5f * ah;
      const float ax2 = ax + 0.5f * aw, ay2 = ay + 0.5f * ah;
      const float aarea = aw * ah;
      // best IoU vs valid truths (initial 0); early exit once >= THRESH
      float best = 0.f;
      for (int t = 0; t < nv; ++t) {
        float iw = fminf(ax2, bx2[t]) - fmaxf(ax1, bx1[t]);
        float ih = fminf(ay2, by2[t]) - fmaxf(ay1, by1[t]);
        if (iw < 0.f || ih < 0.f) continue;
        float inter = iw * ih;
        float iou = inter / (aarea + barea[t] - inter);
        best = fmaxf(best, iou);
        if (best >= THRESH) break;   // only compared vs THRESH below
      }
      if (best < THRESH) ++cnt;
    }
    noobj_acc += (float)cnt * obj * obj;   // NOOBJECT_SCALE = 1
  }

  float tn = block_reduce_sum(noobj_acc, red);
  if (tid == 0) ws_noobj[b] = tn;
  float tm = block_reduce_sum(warm_acc, red);
  if (tid == 0) ws_warm[b] = tm;
}

// ---------------------------------------------------------------------------
// Kernel B: one block per image (64 thr). Per-truth terms: obj + class loss
// partials, and base/sfac/vb written to workspace for the serial scan.
// ---------------------------------------------------------------------------
__global__ void __launch_bounds__(64)
k_truth(const float* __restrict__ x, const float* __restrict__ truth,
        const float* __restrict__ anchors,
        float* __restrict__ ws_objcls,
        float* __restrict__ ws_base, float* __restrict__ ws_sfac,
        float* __restrict__ ws_vb) {
  __shared__ float s_tw[NT];
  __shared__ int   s_nv;
  __shared__ float red[4];

  const int b   = blockIdx.x;
  const int tid = threadIdx.x;

  if (tid < NT) s_tw[tid] = truth[(size_t)(b * NT + tid) * 5 + 2];
  __syncthreads();
  if (tid == 0) {
    int nv = 0;
    while (nv < NT && s_tw[nv] >= 1e-5f) ++nv;
    s_nv = nv;
  }
  __syncthreads();
  const int nv = s_nv;

  float acc = 0.f;
  if (tid < NT) {
    const int t = tid;
    const float* tr = truth + (size_t)(b * NT + t) * 5;
    const float txv = tr[0], tyv = tr[1], twv = tr[2], thv = tr[3], tcv = tr[4];
    const bool vb = (t < nv);

    int ti = (int)floorf(txv * (float)WW); ti = min(max(ti, 0), WW - 1);
    int tj = (int)floorf(tyv * (float)HH); tj = min(max(tj, 0), HH - 1);
    const int cofs = tj * WW + ti;
    const float* xb = x + (size_t)b * NUM * AL * HW;

    float pw[NUM], ph[NUM];
    for (int n = 0; n < NUM; ++n) {
      pw[n] = xb[(n * AL + 2) * HW + cofs];
      ph[n] = xb[(n * AL + 3) * HW + cofs];
    }

    // shifted IoU (0,0,pw,ph) vs (0,0,tw,th); first-max argmax
    int bn = 0; float bi = -INFINITY;
    for (int n = 0; n < NUM; ++n) {
      float iou = box_iou(0.f, 0.f, pw[n], ph[n], 0.f, 0.f, twv, thv);
      if (iou > bi) { bi = iou; bn = n; }
    }
    const float pwb = pw[bn], phb = ph[bn];

    const float sw = vb ? twv : 1.0f;
    const float sh = vb ? thv : 1.0f;
    const float twt = logf(sw * (float)WW / anchors[2 * (NUM - 1)]);
    const float tht = logf(sh * (float)HH / anchors[2 * (NUM - 1) + 1]);
    const float fi = (float)ti, fj = (float)tj;
    const float dw = twt - pwb, dh = tht - phb;
    float bs = fi * fi + fj * fj + dw * dw + dh * dh;
    bs = vb ? bs : 0.f;

    ws_base[b * NT + t] = bs;
    ws_sfac[b * NT + t] = 2.f - twv * thv;
    ws_vb  [b * NT + t] = vb ? 1.f : 0.f;

    // obj loss ((1 - obj_cell) * 5)^2
    const float obj = xb[4 * HW + cofs];
    if (vb) { float d = (1.f - obj) * 5.f; acc += d * d; }

    // class loss (onehot - cls_pred)^2, preds from anchor bn
    int ci = (int)tcv; ci = min(max(ci, 0), CLASSES - 1);
    if (vb) {
      for (int c = 0; c < CLASSES; ++c) {
        float pr = xb[(bn * AL + 5 + c) * HW + cofs];
        float d = ((c == ci) ? 1.f : 0.f) - pr;
        acc += d * d;
      }
    }
  }

  float tot = block_reduce_sum(acc, red);
  if (tid == 0) ws_objcls[b] = tot;
}

// ---------------------------------------------------------------------------
// Kernel C: single wave. Sum partials (WMMA reduce), run the inherently
// sequential coord-loss scan, emit the scalar loss.
// ---------------------------------------------------------------------------
__global__ void __launch_bounds__(32)
k_final(const float* __restrict__ ws_noobj, const float* __restrict__ ws_objcls,
        const float* __restrict__ ws_warm,
        const float* __restrict__ ws_base, const float* __restrict__ ws_sfac,
        const float* __restrict__ ws_vb,
        const int* __restrict__ seen, float* __restrict__ out) {
  const int lane = threadIdx.x;
  float sn = 0.f, so = 0.f;
  for (int i = lane; i < BB; i += 32) { sn += ws_noobj[i]; so += ws_objcls[i]; }
  sn = wave_reduce_sum(sn);
  so = wave_reduce_sum(so);

  if (lane == 0) {
    const bool use_warm = (seen[0] < 500);
    float L = 0.f;
    for (int b = 0; b < BB; ++b) {
      L += use_warm ? ws_warm[b] : 0.f;
      const float* bp = ws_base + b * NT;
      const float* sp = ws_sfac + b * NT;
      const float* vp = ws_vb   + b * NT;
      for (int t = 0; t < NT; ++t) {
        if (vp[t] > 0.f) {
          float ls = L * sp[t];
          L = L + bp[t] * ls * ls;
        }
      }
    }
    out[0] = sn + so + L;
  }
}

// ---------------------------------------------------------------------------
extern "C" void kernel_launch(void* const* d_in, const int* in_sizes, int n_in,
                              void* d_out, int out_size, void* d_ws, size_t ws_size,
                              hipStream_t stream) {
  (void)in_sizes; (void)n_in; (void)out_size; (void)ws_size;
  const float* x       = (const float*)d_in[0];
  const float* truth   = (const float*)d_in[1];
  const float* anchors = (const float*)d_in[2];
  const int*   seen    = (const int*)d_in[3];

  float* ws        = (float*)d_ws;
  float* ws_noobj  = ws;                    // [256]
  float* ws_warm   = ws + BB;               // [256]
  float* ws_objcls = ws + 2 * BB;           // [256]
  float* ws_base   = ws + 3 * BB;           // [256*50]
  float* ws_sfac   = ws_base + BB * NT;     // [256*50]
  float* ws_vb     = ws_sfac + BB * NT;     // [256*50]

  k_noobj_warm<<<BB, 384, 0, stream>>>(x, truth, ws_noobj, ws_warm);
  k_truth<<<BB, 64, 0, stream>>>(x, truth, anchors, ws_objcls,
                                 ws_base, ws_sfac, ws_vb);
  k_final<<<1, 32, 0, stream>>>(ws_noobj, ws_objcls, ws_warm,
                                ws_base, ws_sfac, ws_vb, seen, (float*)d_out);
}